// NonMaximaSuppression2d_43267500540174
// MI455X (gfx1250) — compile-verified
//
#include <hip/hip_runtime.h>
#include <hip/hip_bf16.h>
#include <cstdint>

// 3x3 non-maximum suppression, replicate padding, fp32.
// out = x if (x > 0 && x strictly greater than all 8 neighbors) else 0.
//
// Memory-bound (AI ~ 0.07 flop/byte): 128 MiB traffic -> ~5.8us floor at
// 23.3 TB/s. Strategy: LDS halo tiles filled with CDNA5 async global->LDS
// DMA (b128 interior + b32 column halo), b128 LDS reads, b128 stores.
//
// NOTE: the async-load LDS destination MUST be derived from &tile[...] (a
// ptrtoint capture of the LDS object). Computing the offset arithmetically
// lets alias analysis prove `tile` is never written, and the compute-phase
// ds_loads get folded to undef (observed in the round-2 histogram: ds=0).

#define TX 128            // output tile width (floats)
#define TY 16             // output tile height
#define LDSW 136          // padded LDS row stride: halo@3, centers@[4..131], halo@132
#define LDSH (TY + 2)     // 18 padded rows
#define NTHREADS 256      // 8 wave32 waves

__global__ __launch_bounds__(NTHREADS)
void nms2d_kernel(const float* __restrict__ x, float* __restrict__ out,
                  int H, int W) {
    __shared__ float tile[LDSH * LDSW];

    const int t  = threadIdx.x;
    const int x0 = blockIdx.x * TX;
    const int y0 = blockIdx.y * TY;
    const size_t plane = (size_t)blockIdx.z * (size_t)H * (size_t)W;
    const float* src = x + plane;

    // ---- Interior fill: LDSH rows x 32 float4-groups, per-lane async b128
    //      global->LDS DMA. Row clamp (replicate) keeps rows contiguous, so
    //      only gy needs clamping; both src and LDS dest are 16B aligned.
    const int NGROUPS = LDSH * 32;                 // 576 -> 2.25 iterations
    for (int g = t; g < NGROUPS; g += NTHREADS) {
        const int r = g >> 5;                      // padded row 0..17
        const int q = g & 31;                      // float4 group in row
        int gy = y0 - 1 + r; gy = gy < 0 ? 0 : (gy >= H ? H - 1 : gy);
        const float* gp = src + (size_t)gy * (size_t)W + (size_t)(x0 + (q << 2));
        // Low 32 bits of a flat LDS-aperture pointer == wave-relative LDS
        // byte offset; taking the address also captures `tile` so the
        // "memory"-clobbered asm is known to write it.
        const uint32_t lds_off =
            (uint32_t)(size_t)(&tile[r * LDSW + 4 + (q << 2)]);
        asm volatile("global_load_async_to_lds_b128 %0, %1, off"
                     :: "v"(lds_off), "v"((uint64_t)(size_t)gp)
                     : "memory");
    }

    // ---- Column halo: 2 replicate-clamped scalars per padded row.
    if (t < LDSH * 2) {
        const int r    = t >> 1;
        const int side = t & 1;                    // 0 = left, 1 = right
        int gy = y0 - 1 + r; gy = gy < 0 ? 0 : (gy >= H ? H - 1 : gy);
        int gx = side ? (x0 + TX) : (x0 - 1);
        gx = gx < 0 ? 0 : (gx >= W ? W - 1 : gx);
        const float* gp = src + (size_t)gy * (size_t)W + (size_t)gx;
        const uint32_t lds_off =
            (uint32_t)(size_t)(&tile[r * LDSW + (side ? (TX + 4) : 3)]);
        asm volatile("global_load_async_to_lds_b32 %0, %1, off"
                     :: "v"(lds_off), "v"((uint64_t)(size_t)gp)
                     : "memory");
    }

    // Drain this wave's async copies, then sync the workgroup. Every wave
    // waits on its own ASYNCcnt before signaling, so after the barrier the
    // whole tile is visible to all waves.
    asm volatile("s_wait_asynccnt 0" ::: "memory");
    __syncthreads();

    // ---- Compute: each thread produces 2 rows x 4 consecutive pixels.
    const int lyh = t >> 5;          // 0..7  : output row pair
    const int lxg = (t & 31) << 2;   // 0..124: pixel-group column
    const int oy  = lyh << 1;        // first output row in tile

    // 4 padded rows feed 2 output rows; window cols [3+lxg .. 8+lxg].
    float w[4][6];
#pragma unroll
    for (int j = 0; j < 4; ++j) {
        const float* rp = &tile[(oy + j) * LDSW + lxg];
        float4 a = *reinterpret_cast<const float4*>(rp + 4);  // aligned b128
        w[j][0] = rp[3];
        w[j][1] = a.x; w[j][2] = a.y; w[j][3] = a.z; w[j][4] = a.w;
        w[j][5] = rp[8];
    }

    float4 res0, res1;
    float* r0p = &res0.x;
    float* r1p = &res1.x;
#pragma unroll
    for (int k = 0; k < 4; ++k) {
        // Full 3-wide maxima (rows used as top/bottom neighbors).
        float h0 = fmaxf(fmaxf(w[0][k], w[0][k + 1]), w[0][k + 2]);
        float h1 = fmaxf(fmaxf(w[1][k], w[1][k + 1]), w[1][k + 2]);
        float h2 = fmaxf(fmaxf(w[2][k], w[2][k + 1]), w[2][k + 2]);
        float h3 = fmaxf(fmaxf(w[3][k], w[3][k + 1]), w[3][k + 2]);
        // Side maxima (rows used as the center row: exclude the center).
        float s1 = fmaxf(w[1][k], w[1][k + 2]);
        float s2 = fmaxf(w[2][k], w[2][k + 2]);
        // Seed with 0: the reference's zeroed center conv channel makes 0
        // participate in the channel-max.
        float m0 = fmaxf(fmaxf(h0, h2), fmaxf(s1, 0.0f));
        float m1 = fmaxf(fmaxf(h1, h3), fmaxf(s2, 0.0f));
        const float c0 = w[1][k + 1];
        const float c1 = w[2][k + 1];
        r0p[k] = (c0 > m0) ? c0 : 0.0f;
        r1p[k] = (c1 > m1) ? c1 : 0.0f;
    }

    float* dst = out + plane + (size_t)(y0 + oy) * (size_t)W + (size_t)(x0 + lxg);
    *reinterpret_cast<float4*>(dst)     = res0;   // coalesced global_store_b128
    *reinterpret_cast<float4*>(dst + W) = res1;
}

extern "C" void kernel_launch(void* const* d_in, const int* in_sizes, int n_in,
                              void* d_out, int out_size, void* d_ws, size_t ws_size,
                              hipStream_t stream) {
    (void)n_in; (void)out_size; (void)d_ws; (void)ws_size;
    const float* x = (const float*)d_in[0];
    float* out = (float*)d_out;

    const int H = 256, W = 256;                 // reference: (4, 64, 256, 256)
    const int planes = in_sizes[0] / (H * W);   // B*C = 256

    dim3 grid(W / TX, H / TY, planes);          // (2, 16, 256)
    nms2d_kernel<<<grid, NTHREADS, 0, stream>>>(x, out, H, W);
}